// get_model_5196910428875
// MI455X (gfx1250) — compile-verified
//
#include <hip/hip_runtime.h>

// Permutohedral splat for D1=3, S=384. Output: (B, 128, 128, 3) f32.
#define S_DIM 384

typedef float v2f __attribute__((ext_vector_type(2)));
typedef float v8f __attribute__((ext_vector_type(8)));

// Elevate this lane's point p = (p0,p1,p2) by the 3x3 transform T using
// V_WMMA_F32_16X16X4_F32 (D = A(16x4) x B(4x16) + C). Two WMMAs per wave:
// tile A = points on lanes 0-15, tile B = points on lanes 16-31.
// 32-bit A layout: lanes0-15 {V0=K0,V1=K1}, lanes16-31 {V0=K2,V1=K3}, M=lane&15.
// 32-bit B layout (K x N): lanes0-15 {V0=K0,V1=K1} cols, lanes16-31 {V0=K2,V1=K3}.
// D layout: VGPR j holds row M=j (lanes0-15) / M=j+8 (lanes16-31), N=lane&15.
__device__ inline void elevate_wmma(float p0, float p1, float p2,
                                    float t00, float t01, float t02,
                                    float t10, float t11, float t12,
                                    float t20, float t21, float t22,
                                    float& e0, float& e1, float& e2) {
  const int lane = threadIdx.x & 31;
  const int m = lane & 15;
  const bool lo = lane < 16;

  // A[m][k] = T[m][k] for m<3,k<3 else 0
  float axl = (m == 0) ? t00 : (m == 1) ? t10 : (m == 2) ? t20 : 0.f; // T[m][0]
  float axh = (m == 0) ? t02 : (m == 1) ? t12 : (m == 2) ? t22 : 0.f; // T[m][2]
  float ayl = (m == 0) ? t01 : (m == 1) ? t11 : (m == 2) ? t21 : 0.f; // T[m][1]
  v2f a;
  a.x = lo ? axl : axh;
  a.y = lo ? ayl : 0.f;

  // Gather point columns into B layout (col = lane&15).
  float sA0 = __shfl(p0, m, 32),      sA1 = __shfl(p1, m, 32),      sA2 = __shfl(p2, m, 32);
  float sB0 = __shfl(p0, m + 16, 32), sB1 = __shfl(p1, m + 16, 32), sB2 = __shfl(p2, m + 16, 32);

  v2f bA; bA.x = lo ? sA0 : sA2; bA.y = lo ? sA1 : 0.f;
  v2f bB; bB.x = lo ? sB0 : sB2; bB.y = lo ? sB1 : 0.f;

  v8f c = {};
  v8f dA = __builtin_amdgcn_wmma_f32_16x16x4_f32(false, a, false, bA, (short)0, c, false, false);
  v8f dB = __builtin_amdgcn_wmma_f32_16x16x4_f32(false, a, false, bB, (short)0, c, false, false);

  // Rows 0..2 on lanes 0-15 of dA/dB hold elevated coords of the 16 points.
  float f0 = __shfl(dB[0], m, 32), f1 = __shfl(dB[1], m, 32), f2 = __shfl(dB[2], m, 32);
  e0 = lo ? dA[0] : f0;
  e1 = lo ? dA[1] : f1;
  e2 = lo ? dA[2] : f2;
}

// Reference lattice math for one point. Outputs adjusted greedy coords (axes
// 0,1; exact multiples of 3) and ranks clamped to [0,2] (JAX OOB-gather clamp).
// All rs/cond/sign logic is exact-integer: greedy coords are small exact
// multiples of 3, so the reference's float compares on integer-valued floats
// are bit-identical to int compares.
__device__ inline void lattice_point(float e0, float e1, float e2,
                                     int& g0, int& g1, int& rc0, int& rc1) {
  // jnp.round == round-half-to-even == rintf (RNE default mode).
  // Keep the correctly-rounded /3.0f to match the reference's rounding cuts.
  float gr0 = rintf(e0 / 3.0f) * 3.0f;
  float gr1 = rintf(e1 / 3.0f) * 3.0f;
  float gr2 = rintf(e2 / 3.0f) * 3.0f;
  float m0 = e0 - gr0, m1 = e1 - gr1, m2 = e2 - gr2;
  // rank = argsort(argsort(-emg, stable)): position in descending order,
  // ties broken by original index.
  int r0 = (int)(m1 > m0) + (int)(m2 > m0);
  int r1 = (int)(m0 > m1) + (int)(m2 > m1) + (int)(m0 == m1);
  int gi0 = (int)gr0, gi1 = (int)gr1, gi2 = (int)gr2; // exact multiples of 3
  int rsi = (gi0 + gi1 + gi2) / 3;                    // exact (sum % 3 == 0)
  int step = (rsi > 0) ? -3 : ((rsi < 0) ? 3 : 0);
  bool c0 = (rsi > 0) ? (r0 >= 3 - rsi) : ((rsi < 0) ? (r0 < -rsi) : false);
  bool c1 = (rsi > 0) ? (r1 >= 3 - rsi) : ((rsi < 0) ? (r1 < -rsi) : false);
  g0 = gi0 + (c0 ? step : 0);
  g1 = gi1 + (c1 ? step : 0);
  r0 += (c0 ? step : 0) + rsi;
  r1 += (c1 ? step : 0) + rsi;
  rc0 = min(max(r0, 0), 2);
  rc1 = min(max(r1, 0), 2);
}

__global__ __launch_bounds__(256) void pass1_minkeys(const float* __restrict__ pc,
                                                     const float* __restrict__ tm,
                                                     int* __restrict__ mins, int N) {
  const int b = blockIdx.y;
  int n = blockIdx.x * 256 + threadIdx.x;
  const bool valid = n < N;
  n = valid ? n : (N - 1);

  const float* T = tm + (size_t)b * 9;
  float t00 = T[0], t01 = T[1], t02 = T[2];
  float t10 = T[3], t11 = T[4], t12 = T[5];
  float t20 = T[6], t21 = T[7], t22 = T[8];

  const size_t base = (size_t)b * 3 * (size_t)N;
  float p0 = pc[base + n];
  float p1 = pc[base + (size_t)N + n];
  float p2 = pc[base + 2 * (size_t)N + n];

  float e0, e1, e2;
  elevate_wmma(p0, p1, p2, t00, t01, t02, t10, t11, t12, t20, t21, t22, e0, e1, e2);

  int g0, g1, rc0, rc1;
  lattice_point(e0, e1, e2, g0, g1, rc0, rc1);

  // offset[a] = min_n (g_a - clamped_rank_a)  (min over all simplex vertices)
  int v0 = g0 - rc0;
  int v1 = g1 - rc1;
#pragma unroll
  for (int off = 16; off; off >>= 1) {
    v0 = min(v0, __shfl_xor(v0, off, 32));
    v1 = min(v1, __shfl_xor(v1, off, 32));
  }
  if ((threadIdx.x & 31) == 0) {
    atomicMin(&mins[b * 2 + 0], v0);
    atomicMin(&mins[b * 2 + 1], v1);
  }
}

__global__ __launch_bounds__(256) void pass2_splat(const float* __restrict__ pc,
                                                   const float* __restrict__ feat,
                                                   const float* __restrict__ tm,
                                                   const int* __restrict__ mins,
                                                   float* __restrict__ out, int N) {
  const int b = blockIdx.y;
  int n = blockIdx.x * 256 + threadIdx.x;
  const bool valid = n < N;
  n = valid ? n : (N - 1);

  const int off0 = mins[b * 2 + 0];
  const int off1 = mins[b * 2 + 1];
  const int pick0 = (((-off0) % 3) + 3) % 3;
  const int pick1 = (((-off1) % 3) + 3) % 3;

  const float* T = tm + (size_t)b * 9;
  float t00 = T[0], t01 = T[1], t02 = T[2];
  float t10 = T[3], t11 = T[4], t12 = T[5];
  float t20 = T[6], t21 = T[7], t22 = T[8];

  const size_t base = (size_t)b * 3 * (size_t)N;
  float p0 = pc[base + n];
  float p1 = pc[base + (size_t)N + n];
  float p2 = pc[base + 2 * (size_t)N + n];

  float e0, e1, e2;
  elevate_wmma(p0, p1, p2, t00, t01, t02, t10, t11, t12, t20, t21, t22, e0, e1, e2);

  int g0, g1, rc0, rc1;
  lattice_point(e0, e1, e2, g0, g1, rc0, rc1);

  const int c0 = g0 - off0; // >= 0 and == pick0 (mod 3) by construction
  const int c1 = g1 - off1;
  if (valid && c0 < S_DIM && c1 < S_DIM) {
    const int i = (c0 - pick0) / 3; // 0..127
    const int k = (c1 - pick1) / 3; // 0..127
    float f0 = feat[base + n];
    float f1 = feat[base + (size_t)N + n];
    float f2 = feat[base + 2 * (size_t)N + n];
    float* o = out + (((size_t)b * 128 + (size_t)i) * 128 + (size_t)k) * 3;
    __hip_atomic_fetch_add(o + 0, f0, __ATOMIC_RELAXED, __HIP_MEMORY_SCOPE_AGENT);
    __hip_atomic_fetch_add(o + 1, f1, __ATOMIC_RELAXED, __HIP_MEMORY_SCOPE_AGENT);
    __hip_atomic_fetch_add(o + 2, f2, __ATOMIC_RELAXED, __HIP_MEMORY_SCOPE_AGENT);
  }
}

extern "C" void kernel_launch(void* const* d_in, const int* in_sizes, int n_in,
                              void* d_out, int out_size, void* d_ws, size_t ws_size,
                              hipStream_t stream) {
  const float* pc   = (const float*)d_in[0]; // (B,3,N)
  const float* feat = (const float*)d_in[1]; // (B,3,N)
  const float* tm   = (const float*)d_in[2]; // (B,3,3)
  float* out = (float*)d_out;                // (B,128,128,3)
  int* mins = (int*)d_ws;                    // 2*B ints

  const int B = in_sizes[2] / 9;
  const int N = in_sizes[0] / (3 * B);

  // Zero the output grid; fill per-batch mins with 0x7F7F7F7F (~INT_MAX).
  hipMemsetAsync(d_out, 0, (size_t)out_size * sizeof(float), stream);
  hipMemsetAsync(d_ws, 0x7F, (size_t)(2 * B) * sizeof(int), stream);

  dim3 grid((N + 255) / 256, B);
  pass1_minkeys<<<grid, 256, 0, stream>>>(pc, tm, mins, N);
  pass2_splat<<<grid, 256, 0, stream>>>(pc, feat, tm, mins, out, N);
}